// MTL_66537633350042
// MI455X (gfx1250) — compile-verified
//
#include <hip/hip_runtime.h>
#include <hip/hip_bf16.h>
#include <math.h>

// ---------------------------------------------------------------------------
// CDNA5 (gfx1250) BTSP step, fp32.  Matvecs via V_WMMA_F32_16X16X4_F32:
// wave = 16 rows, K stepped by 8 with one b128 load per lane and TWO WMMAs
// into independent accumulators (hides the WMMA->WMMA RAW hazard).  K-slot
// binding: lane half h holds k+4h..k+4h+3; WMMA#1 consumes regs {0,1} of the
// b128 (k+4h, k+4h+1), WMMA#2 regs {2,3}.  B column 0 (N=0 -> lanes 0 and 16)
// carries the matching x values, so each slot's A/B pair is consistent and
// D column 0 accumulates the exact row dots.
// ---------------------------------------------------------------------------

typedef float v2f __attribute__((ext_vector_type(2)));
typedef float v4f __attribute__((ext_vector_type(4)));
typedef float v8f __attribute__((ext_vector_type(8)));

// Order-preserving float <-> uint mapping for radix selection.
__device__ __forceinline__ unsigned f2k(float f) {
  unsigned u = __float_as_uint(f);
  return (u & 0x80000000u) ? ~u : (u | 0x80000000u);
}
__device__ __forceinline__ float k2f(unsigned k) {
  unsigned u = (k & 0x80000000u) ? (k ^ 0x80000000u) : ~k;
  return __uint_as_float(u);
}

// ---------------------------------------------------------------------------
// part[chunk*R + row] = dot(W[row, k0:k0+kLen], x[k0:k0+kLen])
// Deterministic split-K: each blockIdx.y owns a disjoint K-range + slice.
// ---------------------------------------------------------------------------
__global__ void mv_wmma_f32(const float* __restrict__ W,
                            const float* __restrict__ x,
                            float* __restrict__ part,
                            int R, int C, int kLen) {
  const int lane          = threadIdx.x & 31;
  const int wave          = threadIdx.x >> 5;
  const int wavesPerBlock = blockDim.x >> 5;
  const int rowBase       = (blockIdx.x * wavesPerBlock + wave) * 16;
  if (rowBase >= R) return;                  // wave-uniform exit

  const int laneHalf = lane >> 4;
  const int laneRow  = lane & 15;
  const bool isN0    = (laneRow == 0);       // lanes 0 and 16: column N=0
  const int k0       = blockIdx.y * kLen;
  const float* __restrict__ wrow = W + (size_t)(rowBase + laneRow) * (size_t)C;

  v8f acc0 = {0.f,0.f,0.f,0.f,0.f,0.f,0.f,0.f};
  v8f acc1 = {0.f,0.f,0.f,0.f,0.f,0.f,0.f,0.f};
  for (int k = k0; k < k0 + kLen; k += 8) {
    const int ka = k + laneHalf * 4;
    const float* wp = wrow + ka;
    const v4f wv = *(const v4f*)wp;                    // 16B/lane A stream
    const v4f xv = *(const v4f*)(x + ka);
    __builtin_prefetch(wp + 128, 0, 0);                // 512B ahead (spec.)
    v2f a0; a0.x = wv.x; a0.y = wv.y;
    v2f a1; a1.x = wv.z; a1.y = wv.w;
    v2f b0; b0.x = isN0 ? xv.x : 0.0f; b0.y = isN0 ? xv.y : 0.0f;
    v2f b1; b1.x = isN0 ? xv.z : 0.0f; b1.y = isN0 ? xv.w : 0.0f;
    acc0 = __builtin_amdgcn_wmma_f32_16x16x4_f32(false, a0, false, b0,
                                                 (short)0, acc0, false, false);
    acc1 = __builtin_amdgcn_wmma_f32_16x16x4_f32(false, a1, false, b1,
                                                 (short)0, acc1, false, false);
  }
  const v8f acc = acc0 + acc1;
  // D(M=m, N=0): lane 0 -> M=0..7, lane 16 -> M=8..15, VGPRs 0..7.
  if (isN0) {
    float* p = part + (size_t)blockIdx.y * (size_t)R + rowBase + laneHalf * 8;
#pragma unroll
    for (int m = 0; m < 8; ++m) p[m] = acc[m];
  }
}

// ---------------------------------------------------------------------------
// Fused single pass over W_ca3_ca1 (256MB read once):
//   dots:   part += W[row,:] . x_ca3                  (WMMA)
//   write:  Wn[row,j] = (1-a*IS[row])*W[row,j] + a*IS[row]*x_ca3[j]  (b128)
// ---------------------------------------------------------------------------
__global__ void fused_btsp(const float* __restrict__ W,
                           const float* __restrict__ xca3,
                           const float* __restrict__ IS,
                           float* __restrict__ Wn,
                           float* __restrict__ part,
                           int R, int C, int kLen, float alpha) {
  const int lane          = threadIdx.x & 31;
  const int wave          = threadIdx.x >> 5;
  const int wavesPerBlock = blockDim.x >> 5;
  const int rowBase       = (blockIdx.x * wavesPerBlock + wave) * 16;
  if (rowBase >= R) return;

  const int laneHalf = lane >> 4;
  const int laneRow  = lane & 15;
  const bool isN0    = (laneRow == 0);
  const int row      = rowBase + laneRow;
  const int k0       = blockIdx.y * kLen;

  const float isr  = IS[row];
  const float keep = 1.0f - alpha * isr;
  const float wr   = alpha * isr;

  const float* __restrict__ wrow = W  + (size_t)row * (size_t)C;
  float* __restrict__       nrow = Wn + (size_t)row * (size_t)C;

  v8f acc0 = {0.f,0.f,0.f,0.f,0.f,0.f,0.f,0.f};
  v8f acc1 = {0.f,0.f,0.f,0.f,0.f,0.f,0.f,0.f};
  for (int k = k0; k < k0 + kLen; k += 8) {
    const int ka = k + laneHalf * 4;
    const float* wp = wrow + ka;
    const v4f wv = *(const v4f*)wp;
    const v4f xv = *(const v4f*)(xca3 + ka);           // real x for update
    __builtin_prefetch(wp + 128, 0, 0);
    v2f a0; a0.x = wv.x; a0.y = wv.y;
    v2f a1; a1.x = wv.z; a1.y = wv.w;
    v2f b0; b0.x = isN0 ? xv.x : 0.0f; b0.y = isN0 ? xv.y : 0.0f;
    v2f b1; b1.x = isN0 ? xv.z : 0.0f; b1.y = isN0 ? xv.w : 0.0f;
    acc0 = __builtin_amdgcn_wmma_f32_16x16x4_f32(false, a0, false, b0,
                                                 (short)0, acc0, false, false);
    acc1 = __builtin_amdgcn_wmma_f32_16x16x4_f32(false, a1, false, b1,
                                                 (short)0, acc1, false, false);
    v4f nv;                                            // rank-1 BTSP write
    nv.x = fmaf(keep, wv.x, wr * xv.x);
    nv.y = fmaf(keep, wv.y, wr * xv.y);
    nv.z = fmaf(keep, wv.z, wr * xv.z);
    nv.w = fmaf(keep, wv.w, wr * xv.w);
    *(v4f*)(nrow + ka) = nv;                           // b128 store
  }
  const v8f acc = acc0 + acc1;
  if (isN0) {
    float* p = part + (size_t)blockIdx.y * (size_t)R + rowBase + laneHalf * 8;
#pragma unroll
    for (int m = 0; m < 8; ++m) p[m] = acc[m];
  }
}

// y[i] = sum_c part[c*R + i]   (fixed order -> deterministic)
__global__ void reduce_part(const float* __restrict__ part,
                            float* __restrict__ y, int R, int chunks) {
  const int i = blockIdx.x * blockDim.x + threadIdx.x;
  if (i >= R) return;
  float s = 0.0f;
  for (int c = 0; c < chunks; ++c) s += part[(size_t)c * (size_t)R + i];
  y[i] = s;
}

// ---------------------------------------------------------------------------
// Exact kth-largest via 32-pass MSB radix select on order-preserving keys
// (single 1024-thread block, keys in LDS), then sigmoid(beta*(v-thr)).
// ---------------------------------------------------------------------------
__global__ void topk_sigmoid(const float* __restrict__ y,
                             const float* __restrict__ bias,
                             float* __restrict__ out,
                             int n, int K, float beta) {
  __shared__ unsigned keys[8192];           // 32KB of the 320KB WGP LDS
  __shared__ unsigned s_cnt;
  const int t = threadIdx.x, nt = blockDim.x;

  for (int i = t; i < n; i += nt) {
    const float v = y[i] + (bias ? bias[i] : 0.0f);
    keys[i] = f2k(v);
  }
  __syncthreads();

  unsigned prefix = 0;
  int remaining = K;
  for (int bit = 31; bit >= 0; --bit) {
    if (t == 0) s_cnt = 0;
    __syncthreads();
    const unsigned want = prefix | (1u << bit);
    const unsigned mask = ~((1u << bit) - 1u);   // bits 31..bit
    unsigned local = 0;
    for (int i = t; i < n; i += nt)
      local += ((keys[i] & mask) == want) ? 1u : 0u;
    if (local) atomicAdd(&s_cnt, local);
    __syncthreads();
    const int c = (int)s_cnt;
    if (c >= remaining) prefix = want;           // kth-largest has this bit
    else                remaining -= c;
    __syncthreads();
  }

  const float thr = k2f(prefix);                 // exact kth-largest value
  for (int i = t; i < n; i += nt) {
    const float v = y[i] + (bias ? bias[i] : 0.0f);
    out[i] = 1.0f / (1.0f + __expf(-beta * (v - thr)));
  }
}

// ---------------------------------------------------------------------------
extern "C" void kernel_launch(void* const* d_in, const int* in_sizes, int n_in,
                              void* d_out, int out_size, void* d_ws, size_t ws_size,
                              hipStream_t stream) {
  (void)in_sizes; (void)n_in; (void)out_size; (void)ws_size;

  const int DIM_EI = 4096, DIM_CA1 = 8192, DIM_CA3 = 8192, DIM_EO = 4096;
  const float BETA = 10.0f, BETA_CA3 = 1000.0f, ALPHA = 0.01f;
  const int K_CA3 = 64, K_LAT = 128, K_OUT = 64;

  const float* x_ei      = (const float*)d_in[0];
  const float* W_ei_ca3  = (const float*)d_in[1];
  const float* W_ei_ca1  = (const float*)d_in[2];
  const float* W_ca3_ca1 = (const float*)d_in[3];
  const float* W_ca1_eo  = (const float*)d_in[4];
  const float* B_ei_ca1  = (const float*)d_in[5];
  const float* B_ca1_eo  = (const float*)d_in[6];

  float* out      = (float*)d_out;
  float* xeo_out  = out;               // [4096]
  float* Wnew_out = out + DIM_EO;      // [8192*8192]

  float* ws    = (float*)d_ws;         // fp32 workspace layout:
  float* y1    = ws;                   // 8192  W_ei_ca3 @ x_ei
  float* x_ca3 = ws + 8192;            // 8192
  float* y3    = ws + 16384;           // 8192  W_ei_ca1 @ x_ei
  float* IS    = ws + 24576;           // 8192
  float* y2    = ws + 32768;           // 8192  W_ca3_ca1 @ x_ca3
  float* x_ca1 = ws + 40960;           // 8192
  float* y4    = ws + 49152;           // 4096  W_ca1_eo @ x_ca1
  float* part  = ws + 53248;           // up to 8*8192 split-K partials

  const dim3 blk128(128);              // 4 wave32s / block
  const dim3 blk256(256);
  const dim3 blk1024(1024);

  // 1) y1 = W_ei_ca3 @ x_ei    (8192x4096, split-K 8)
  mv_wmma_f32<<<dim3(DIM_CA3 / 64, 8), blk128, 0, stream>>>(
      W_ei_ca3, x_ei, part, DIM_CA3, DIM_EI, DIM_EI / 8);
  reduce_part<<<dim3(DIM_CA3 / 256), blk256, 0, stream>>>(part, y1, DIM_CA3, 8);
  // 2) x_ca3 = sparsemoid(y1, K=64, beta_ca3)
  topk_sigmoid<<<dim3(1), blk1024, 0, stream>>>(y1, nullptr, x_ca3,
                                                DIM_CA3, K_CA3, BETA_CA3);

  // 3) IS = sparsemoid(W_ei_ca1 @ x_ei + B, K=128, beta)
  mv_wmma_f32<<<dim3(DIM_CA1 / 64, 8), blk128, 0, stream>>>(
      W_ei_ca1, x_ei, part, DIM_CA1, DIM_EI, DIM_EI / 8);
  reduce_part<<<dim3(DIM_CA1 / 256), blk256, 0, stream>>>(part, y3, DIM_CA1, 8);
  topk_sigmoid<<<dim3(1), blk1024, 0, stream>>>(y3, B_ei_ca1, IS,
                                                DIM_CA1, K_LAT, BETA);

  // 4) Fused: y2 partials = W_ca3_ca1 @ x_ca3 AND W_new write (one 256MB read)
  fused_btsp<<<dim3(DIM_CA1 / 64, 4), blk128, 0, stream>>>(
      W_ca3_ca1, x_ca3, IS, Wnew_out, part, DIM_CA1, DIM_CA3, DIM_CA3 / 4, ALPHA);
  reduce_part<<<dim3(DIM_CA1 / 256), blk256, 0, stream>>>(part, y2, DIM_CA1, 4);
  // 5) x_ca1 = sparsemoid(y2, K=128, beta_ca3)   (pre-update weights: the
  //    fused kernel computed dots from values read before overwriting Wnew)
  topk_sigmoid<<<dim3(1), blk1024, 0, stream>>>(y2, nullptr, x_ca1,
                                                DIM_CA1, K_LAT, BETA_CA3);

  // 6) x_eo = sparsemoid(W_ca1_eo @ x_ca1 + B, K=64, beta)
  mv_wmma_f32<<<dim3(DIM_EO / 64, 8), blk128, 0, stream>>>(
      W_ca1_eo, x_ca1, part, DIM_EO, DIM_CA1, DIM_CA1 / 8);
  reduce_part<<<dim3(DIM_EO / 256), blk256, 0, stream>>>(part, y4, DIM_EO, 8);
  topk_sigmoid<<<dim3(1), blk1024, 0, stream>>>(y4, B_ca1_eo, xeo_out,
                                                DIM_EO, K_OUT, BETA);
}